// NeuralVoxelHash_69776038690999
// MI455X (gfx1250) — compile-verified
//
#include <hip/hip_runtime.h>

typedef float v4f __attribute__((ext_vector_type(4)));

namespace {
constexpr long long kBuf   = 5000000LL;
constexpr unsigned  kBufU  = 5000000u;
constexpr long long kNFeat = 4194304LL;
constexpr int       kDim   = 8;
// P mod B  (B = 5,000,000)
constexpr unsigned  kP0m = 3856093u;   // 73856093 % B
constexpr unsigned  kP1m = 4349669u;   // 19349669 % B
constexpr unsigned  kP2m = 3492791u;   // 83492791 % B
// (256*Pm) mod B
constexpr unsigned  kK1x = 2159808u, kK1y = 3515264u, kK1z = 4154496u;
// (-32768*P) mod B  (floored)
constexpr unsigned  kC0x = 3544576u, kC0y = 46208u,  kC0z = 3224512u;
}

// v % 5,000,000 for v < 2^31.  5e6 = 2^6 * 78125; magic M=ceil(2^47/78125)
// is exact for t < 2^28 (Granlund-Montgomery bound: M*d-2^47 = 4047 < 2^19).
__device__ __forceinline__ unsigned mod5m(unsigned v) {
    unsigned t = v >> 6;                            // < 2^25
    unsigned q = __umulhi(t, 1801439851u) >> 15;    // t / 78125
    unsigned r = t - q * 78125u;                    // t % 78125
    return (r << 6) | (v & 63u);
}

// (ix * P) mod B, floored, all 32-bit.  Valid for |ix| < 32768 (inputs give
// |ix| <= ~500); out-of-range ix still yields an in-range key (memory-safe).
__device__ __forceinline__ unsigned axis_hash(int ix, unsigned K1, unsigned K2,
                                              unsigned C0) {
    unsigned j = (unsigned)(ix + 32768);
    unsigned v = (j >> 8) * K1 + ((j & 255u) * K2 + C0);   // < 2^31
    return mod5m(v);
}

// (a + b) mod B for a,b < B — unsigned-wrap min trick
__device__ __forceinline__ unsigned add_mod(unsigned a, unsigned b) {
    unsigned t = a + b;
    return min(t, t - kBufU);
}
// (a + b + c) mod B for a,b,c < B
__device__ __forceinline__ unsigned sum3_mod(unsigned a, unsigned b, unsigned c) {
    unsigned t = a + b + c;                 // < 3B < 2^24*... (1.5e7)
    t = min(t, t - 2u * kBufU);
    return min(t, t - kBufU);
}

__global__ __launch_bounds__(256)
void voxel_hash_interp(const float* __restrict__ qp,
                       const float* __restrict__ feats,
                       const long long* __restrict__ itab,
                       float* __restrict__ out, int n)
{
    int i = blockIdx.x * blockDim.x + threadIdx.x;
    if (i >= n) return;

    const float px = qp[3 * i + 0];   // coalesced global_load_b96
    const float py = qp[3 * i + 1];
    const float pz = qp[3 * i + 2];

    // One IEEE division per axis; levels 1,2 are exact 2^-k rescales
    // (0.6f==2*0.3f, 1.2f==4*0.3f; rounding commutes with 2^k scaling).
    const float tX = px / 0.3f, tY = py / 0.3f, tZ = pz / 0.3f;

    v4f accL = {0.f, 0.f, 0.f, 0.f};
    v4f accH = {0.f, 0.f, 0.f, 0.f};

#pragma unroll
    for (int lvl = 0; lvl < 3; ++lvl) {
        const float scale = (lvl == 0) ? 1.0f : (lvl == 1 ? 0.5f : 0.25f);
        const float tx = tX * scale, ty = tY * scale, tz = tZ * scale;
        const float bx = floorf(tx), by = floorf(ty), bz = floorf(tz);
        const float dx = tx - bx, dy = ty - by, dz = tz - bz;

        // Per-axis floored hashes mod B, fully 32-bit.
        const unsigned ax = axis_hash((int)bx, kK1x, kP0m, kC0x);
        const unsigned ay = axis_hash((int)by, kK1y, kP1m, kC0y);
        const unsigned az = axis_hash((int)bz, kK1z, kP2m, kC0z);

        // Corner keys: one full sum, then 7 modular increments by Pm.
        // Order == STEPS rows: (x,y,z) bits, z fastest.
        unsigned key[8];
        key[0] = sum3_mod(ax, ay, az);        // 000
        key[1] = add_mod(key[0], kP2m);       // 001
        key[2] = add_mod(key[0], kP1m);       // 010
        key[3] = add_mod(key[2], kP2m);       // 011
        key[4] = add_mod(key[0], kP0m);       // 100
        key[5] = add_mod(key[4], kP2m);       // 101
        key[6] = add_mod(key[4], kP1m);       // 110
        key[7] = add_mod(key[6], kP2m);       // 111

        // Index gathers: uniform SGPR base + 32-bit voffset, b64 clause.
        const char* __restrict__ tabB =
            (const char*)(itab + (long long)lvl * kBuf);
        int2 e[8];
#pragma unroll
        for (int c = 0; c < 8; ++c)
            e[c] = *(const int2*)(tabB + (size_t)(key[c] * 8u));

        // valid iff all int64 idx >= 0  <=>  OR of high words non-negative
        const int orHi = e[0].y | e[1].y | e[2].y | e[3].y |
                         e[4].y | e[5].y | e[6].y | e[7].y;
        const float vm = (orHi >= 0) ? 1.f : 0.f;

        // Fold mask into x-axis weights (covers all 8 corner weights).
        const float wx0 = (1.f - dx) * vm, wx1 = dx * vm;
        const float wy0 = 1.f - dy, wy1 = dy;
        const float wz0 = 1.f - dz, wz1 = dz;
        const float wyz[4] = { wy0 * wz0, wy0 * wz1, wy1 * wz0, wy1 * wz1 };
        const float w[8] = { wx0*wyz[0], wx0*wyz[1], wx0*wyz[2], wx0*wyz[3],
                             wx1*wyz[0], wx1*wyz[1], wx1*wyz[2], wx1*wyz[3] };

        const char* __restrict__ fB =
            (const char*)(feats + (long long)lvl * kNFeat * (long long)kDim);
#pragma unroll
        for (int c = 0; c < 8; ++c) {
            // Clamp row id in-bounds; invalid levels are zeroed by vm anyway,
            // and valid ids are < NFEAT so min() is a no-op -> output-exact.
            const unsigned off = min((unsigned)e[c].x, 4194303u) << 5;
            const v4f lo = *(const v4f*)(fB + off);        // global_load_b128
            const v4f hi = *(const v4f*)(fB + off + 16u);  // global_load_b128
            const v4f wv = {w[c], w[c], w[c], w[c]};
            accL = __builtin_elementwise_fma(lo, wv, accL);
            accH = __builtin_elementwise_fma(hi, wv, accH);
        }
    }

    v4f* __restrict__ o = (v4f*)(out + (long long)i * kDim);
    o[0] = accL;   // global_store_b128
    o[1] = accH;   // global_store_b128
}

extern "C" void kernel_launch(void* const* d_in, const int* in_sizes, int n_in,
                              void* d_out, int out_size, void* d_ws, size_t ws_size,
                              hipStream_t stream)
{
    const float*     qp    = (const float*)d_in[0];     // [N,3] f32
    const float*     feats = (const float*)d_in[1];     // [3, 4194304, 8] f32
    const long long* itab  = (const long long*)d_in[2]; // [3, 5000000] i64
    float* out = (float*)d_out;                         // [N,8] f32

    const int n = in_sizes[0] / 3;
    const int block = 256;                              // 8 wave32 per block
    const int grid  = (n + block - 1) / block;
    voxel_hash_interp<<<grid, block, 0, stream>>>(qp, feats, itab, out, n);
}